// FixedNet_10496900072251
// MI455X (gfx1250) — compile-verified
//
#include <hip/hip_runtime.h>
#include <math.h>

#define N_TOTAL  50000
#define N0       20000
#define N1       15000
#define N2       15000
#define OFF1     20000
#define OFF2     35000
#define NEDGE    800000
#define DHID     128
#define IN_DIM   512
#define NCLS     8

typedef __attribute__((ext_vector_type(2))) float v2f;
typedef __attribute__((ext_vector_type(8))) float v8f;

// ---------------------------------------------------------------- utilities
__global__ void zero_f32(float* __restrict__ p, int n) {
    int i = blockIdx.x * blockDim.x + threadIdx.x;
    int stride = gridDim.x * blockDim.x;
    for (; i < n; i += stride) p[i] = 0.0f;
}

__global__ void deg_kernel(const int* __restrict__ dst, float* __restrict__ deg, int nE) {
    int i = blockIdx.x * blockDim.x + threadIdx.x;
    if (i < nE) atomicAdd(&deg[dst[i]], 1.0f);
}

// One wave per edge (grid-stride). Each lane moves 4 floats of the 128-float row.
// src >= srcLimit rows are exactly `fallback` (e.g. b_op for the gcn-op pass).
__global__ __launch_bounds__(256)
void edge_scatter(const float* __restrict__ x, const float* __restrict__ fallback,
                  int srcLimit, const int* __restrict__ src, const int* __restrict__ dst,
                  float* __restrict__ agg, int nE) {
    int gwave  = (blockIdx.x * blockDim.x + threadIdx.x) >> 5;
    int nwaves = (gridDim.x * blockDim.x) >> 5;
    int lane   = threadIdx.x & 31;
    for (int e = gwave; e < nE; e += nwaves) {
        int s = src[e], d = dst[e];
        const float* xp = (s < srcLimit) ? (x + (size_t)s * DHID) : fallback;
        float4 v = *(const float4*)(xp + lane * 4);
        float* ap = agg + (size_t)d * DHID + lane * 4;
        atomicAdd(ap + 0, v.x);
        atomicAdd(ap + 1, v.y);
        atomicAdd(ap + 2, v.z);
        atomicAdd(ap + 3, v.w);
    }
}

// ------------------------------------------------- fp32 WMMA GEMM
// One wave computes a 16x32 output tile (two 16x16 WMMA accumulators sharing the
// A fragment). Out = A[rows,K] @ W^T + bias  (W is [Nout x K] row-major).
// HAS_SCALE: multiply A rows by 1/max(deg,1).  HAS_ELU: elu on output.
// f32 16x16x4 layout: lane j=lane&15 -> row/col j; lanes 0-15 carry K pair {kb,kb+1},
// lanes 16-31 carry {kb+2,kb+3}. C/D: VGPR r -> rows r and r+8.
template <bool HAS_SCALE, bool HAS_ELU>
__global__ __launch_bounds__(32)
void gemm_wmma_f32(const float* __restrict__ A, int lda,
                   const float* __restrict__ W, int K,
                   const float* __restrict__ bias,
                   const float* __restrict__ rowDeg,
                   float* __restrict__ Out, int ldo,
                   int M) {
    const int lane = threadIdx.x;
    const int half = lane >> 4;     // 0 or 1
    const int j    = lane & 15;
    const int m0   = blockIdx.x * 16;
    const int n0   = blockIdx.y * 32;

    int arow  = m0 + j;
    int arowc = (arow < M) ? arow : (M - 1);          // clamp; masked on store
    const float* ap  = A + (size_t)arowc * lda;
    const float* wp0 = W + (size_t)(n0 + j) * K;      // B column n0+j
    const float* wp1 = W + (size_t)(n0 + 16 + j) * K; // B column n0+16+j
    float scale = 1.0f;
    if (HAS_SCALE) scale = 1.0f / fmaxf(rowDeg[arowc], 1.0f);

    v8f c0, c1;
    #pragma unroll
    for (int r = 0; r < 8; ++r) { c0[r] = 0.0f; c1[r] = 0.0f; }

    for (int kb = 0; kb < K; kb += 4) {
        int kk = kb + 2 * half;
        v2f a;
        a.x = ap[kk];
        a.y = ap[kk + 1];
        if (HAS_SCALE) { a.x *= scale; a.y *= scale; }
        v2f b0, b1;
        b0.x = wp0[kk]; b0.y = wp0[kk + 1];
        b1.x = wp1[kk]; b1.y = wp1[kk + 1];
        c0 = __builtin_amdgcn_wmma_f32_16x16x4_f32(false, a, false, b0,
                                                   (short)0, c0, false, false);
        c1 = __builtin_amdgcn_wmma_f32_16x16x4_f32(false, a, false, b1,
                                                   (short)0, c1, false, false);
    }

    float bn0 = bias[n0 + j];
    float bn1 = bias[n0 + 16 + j];
    #pragma unroll
    for (int r = 0; r < 8; ++r) {
        int row = m0 + r + 8 * half;
        if (row < M) {
            float v0 = c0[r] + bn0;
            float v1 = c1[r] + bn1;
            if (HAS_ELU) {
                v0 = (v0 > 0.0f) ? v0 : (expf(v0) - 1.0f);
                v1 = (v1 > 0.0f) ? v1 : (expf(v1) - 1.0f);
            }
            Out[(size_t)row * ldo + n0 + j]      = v0;
            Out[(size_t)row * ldo + n0 + 16 + j] = v1;
        }
    }
}

// ------------------- cluster-select combine: h = h0 + (sel ? one_hot_h : gcn_out)
__global__ void combine_kernel(float* __restrict__ h, const float* __restrict__ agg,
                               const float* __restrict__ deg,
                               const float* __restrict__ We1, const float* __restrict__ be1,
                               const float* __restrict__ We2, const float* __restrict__ be2,
                               const int* __restrict__ assign) {
    int idx = blockIdx.x * blockDim.x + threadIdx.x;
    if (idx >= N_TOTAL * DHID) return;
    int i = idx >> 7;
    int d = idx & 127;
    int na = assign[i];
    bool sel = (na == 0) || (na == 3);     // CLUSTER_IS_ONEHOT = [T,F,F,T]
    float h0v = (i < N0) ? h[idx] : 0.0f;
    float val;
    if (sel) {
        if (i < N0)        val = 0.0f;
        else if (i < OFF2) val = We1[(size_t)d * N1 + (i - OFF1)] + be1[d];
        else               val = We2[(size_t)d * N2 + (i - OFF2)] + be2[d];
    } else {
        val = agg[idx] / fmaxf(deg[i], 1.0f);
    }
    h[idx] = h0v + val;
}

// ------------------------------------------ classifier head (writes logits twice)
__global__ void logits_kernel(const float* __restrict__ ne, const float* __restrict__ Wc,
                              const float* __restrict__ bc,
                              float* __restrict__ o1, float* __restrict__ o2) {
    int t = blockIdx.x * blockDim.x + threadIdx.x;
    if (t >= N_TOTAL * NCLS) return;
    int i = t >> 3, c = t & 7;
    const float* np = ne + (size_t)i * DHID;
    const float* wp = Wc + c * DHID;
    float s = bc[c];
    for (int k = 0; k < DHID; k += 4) {
        float4 a = *(const float4*)(np + k);
        float4 w = *(const float4*)(wp + k);
        s += a.x * w.x + a.y * w.y + a.z * w.z + a.w * w.w;
    }
    o1[t] = s;
    o2[t] = s;
}

// ---------------------------------------------------------------------------
extern "C" void kernel_launch(void* const* d_in, const int* in_sizes, int n_in,
                              void* d_out, int out_size, void* d_ws, size_t ws_size,
                              hipStream_t stream) {
    const float* features0 = (const float*)d_in[0];
    const float* W_pre     = (const float*)d_in[1];
    const float* b_pre     = (const float*)d_in[2];
    const float* W_emb1    = (const float*)d_in[3];
    const float* b_emb1    = (const float*)d_in[4];
    const float* W_emb2    = (const float*)d_in[5];
    const float* b_emb2    = (const float*)d_in[6];
    const float* W_op      = (const float*)d_in[7];
    const float* b_op      = (const float*)d_in[8];
    const float* W_fc      = (const float*)d_in[9];
    const float* b_fc      = (const float*)d_in[10];
    const float* Wg        = (const float*)d_in[11];
    const float* bg        = (const float*)d_in[12];
    const float* Wc        = (const float*)d_in[13];
    const float* bc        = (const float*)d_in[14];
    const int*   edge_src  = (const int*)d_in[15];
    const int*   edge_dst  = (const int*)d_in[16];
    const int*   assign    = (const int*)d_in[17];
    float* out = (float*)d_out;

    const size_t ND = (size_t)N_TOTAL * DHID;
    float* wsA = (float*)d_ws;      // h0 -> h_attributed      [N,D]
    float* wsB = wsA + ND;          // gcn-op input -> h_trans [N,D]
    float* wsC = wsB + ND;          // edge accumulators       [N,D]
    float* deg = wsC + ND;          // in-degree               [N]

    const int NT = DHID / 32;       // N-tiles per GEMM grid (two 16-wide per wave)

    // degrees
    zero_f32<<<64, 256, 0, stream>>>(deg, N_TOTAL);
    deg_kernel<<<(NEDGE + 255) / 256, 256, 0, stream>>>(edge_dst, deg, NEDGE);

    // 1) h_attr = features0 @ W_pre.T + b_pre  -> wsA rows [0, N0)
    gemm_wmma_f32<false, false><<<dim3(N0 / 16, NT), 32, 0, stream>>>(
        features0, IN_DIM, W_pre, IN_DIM, b_pre, nullptr, wsA, DHID, N0);

    // 3a) gcn-op input: wsB[0:N0) = h0 @ W_op.T + b_op  (rows >= N0 are exactly b_op)
    gemm_wmma_f32<false, false><<<dim3(N0 / 16, NT), 32, 0, stream>>>(
        wsA, DHID, W_op, DHID, b_op, nullptr, wsB, DHID, N0);

    // 3b) mean-agg numerator into wsC (b_op fallback for src >= N0)
    zero_f32<<<2048, 256, 0, stream>>>(wsC, (int)ND);
    edge_scatter<<<2048, 256, 0, stream>>>(wsB, b_op, N0, edge_src, edge_dst, wsC, NEDGE);

    // 4) h_attributed = h0 + select(one_hot, gcn)  -> overwrite wsA
    combine_kernel<<<(int)((ND + 255) / 256), 256, 0, stream>>>(
        wsA, wsC, deg, W_emb1, b_emb1, W_emb2, b_emb2, assign);

    // 5) per-type linear -> wsB
    gemm_wmma_f32<false, false><<<dim3((N0 + 15) / 16, NT), 32, 0, stream>>>(
        wsA, DHID, W_fc, DHID, b_fc, nullptr, wsB, DHID, N0);
    gemm_wmma_f32<false, false><<<dim3((N1 + 15) / 16, NT), 32, 0, stream>>>(
        wsA + (size_t)OFF1 * DHID, DHID, W_fc + DHID * DHID, DHID, b_fc + DHID,
        nullptr, wsB + (size_t)OFF1 * DHID, DHID, N1);
    gemm_wmma_f32<false, false><<<dim3((N2 + 15) / 16, NT), 32, 0, stream>>>(
        wsA + (size_t)OFF2 * DHID, DHID, W_fc + 2 * DHID * DHID, DHID, b_fc + 2 * DHID,
        nullptr, wsB + (size_t)OFF2 * DHID, DHID, N2);

    // 6a) mean-agg of h_transform into wsC
    zero_f32<<<2048, 256, 0, stream>>>(wsC, (int)ND);
    edge_scatter<<<2048, 256, 0, stream>>>(wsB, wsB, N_TOTAL, edge_src, edge_dst, wsC, NEDGE);

    // 6b) node_embedding = elu((wsC/deg) @ Wg.T + bg) -> d_out[0 : N*D)
    gemm_wmma_f32<true, true><<<dim3((N_TOTAL + 15) / 16, NT), 32, 0, stream>>>(
        wsC, DHID, Wg, DHID, bg, deg, out, DHID, N_TOTAL);

    // 6c) logits (twice) -> d_out[N*D : N*D + 2*N*8)
    logits_kernel<<<(N_TOTAL * NCLS + 255) / 256, 256, 0, stream>>>(
        out, Wc, bc, out + ND, out + ND + (size_t)N_TOTAL * NCLS);
}